// HAN_62483184222516
// MI455X (gfx1250) — compile-verified
//
#include <hip/hip_runtime.h>
#include <cstdint>

#define NN 50000
#define EE 800000
#define PP 3
#define INS 128
#define HH 4
#define DD 32
#define HDIM 128
#define HIDS 128
#define OUTC 8
#define NEG_SLOPE 0.2f

typedef __attribute__((ext_vector_type(2))) float v2f;
typedef __attribute__((ext_vector_type(8))) float v8f;

// ---- workspace layout (float indices) ----
#define OFF_FEAT 0LL
#define SZ_FEAT  ((long long)PP*NN*HDIM)        // 19,200,000
#define OFF_ACC  (OFF_FEAT + SZ_FEAT)
#define SZ_ACC   ((long long)PP*NN*HDIM)
#define OFF_EL   (OFF_ACC + SZ_ACC)
#define SZ_NH    ((long long)PP*NN*HH)          // 600,000
#define OFF_ER   (OFF_EL + SZ_NH)
#define OFF_MM   (OFF_ER + SZ_NH)               // int-encoded max
#define OFF_DEN  (OFF_MM + SZ_NH)
#define OFF_EXB  (OFF_DEN + SZ_NH)
#define SZ_EXB   ((long long)PP*EE*HH)          // 9,600,000
#define OFF_WSE  (OFF_EXB + SZ_EXB)
#define SZ_WSE   ((long long)PP*NN)
#define OFF_RED  (OFF_WSE + SZ_WSE)             // [0..2]=wsum  [4..6]=beta
#define WS_END   (OFF_RED + 16)

// order-preserving float<->int encode (involution) for atomicMax on signed int
__device__ __forceinline__ int enc_f(float f) {
    int i = __float_as_int(f);
    return i >= 0 ? i : (i ^ 0x7FFFFFFF);
}
__device__ __forceinline__ float dec_i(int i) {
    return __int_as_float(i >= 0 ? i : (i ^ 0x7FFFFFFF));
}

// ---------------- init ----------------
__global__ void k_init(float* ws) {
    long long stride = (long long)gridDim.x * blockDim.x;
    for (long long i = OFF_ACC + blockIdx.x * (long long)blockDim.x + threadIdx.x;
         i < WS_END; i += stride) {
        if (i >= OFF_MM && i < OFF_MM + SZ_NH)
            ((int*)ws)[i] = (int)0x80000000;   // INT_MIN (encoded -inf identity)
        else
            ws[i] = 0.0f;
    }
}

// ------------- feat = h @ W[p]  (WMMA f32 16x16x4) -------------
__global__ void k_gemm_feat(const float* __restrict__ h,
                            const float* __restrict__ W,
                            float* __restrict__ feat) {
    const int lane = threadIdx.x & 31;
    const int wid  = threadIdx.x >> 5;
    const int gid  = blockIdx.x * 8 + wid;     // wave-uniform
    const int MT = NN / 16, NT = HDIM / 16;    // 3125, 8
    if (gid >= PP * MT * NT) return;           // uniform per wave; EXEC stays full
    const int p  = gid / (MT * NT);
    const int r  = gid % (MT * NT);
    const int mt = r / NT, nt = r % NT;
    const int g  = lane >> 4;                  // half-wave: K pair select
    const int t  = lane & 15;

    const float* A = h + (long long)(mt * 16 + t) * INS;
    const float* B = W + (long long)p * INS * HDIM + nt * 16 + t;

    v8f c = {};
    #pragma unroll 4
    for (int k = 0; k < INS; k += 4) {
        v2f a, b;
        a.x = A[k + 2 * g];
        a.y = A[k + 2 * g + 1];
        b.x = B[(long long)(k + 2 * g) * HDIM];
        b.y = B[(long long)(k + 2 * g + 1) * HDIM];
        c = __builtin_amdgcn_wmma_f32_16x16x4_f32(false, a, false, b,
                                                  (short)0, c, false, false);
    }
    float* C = feat + (long long)p * NN * HDIM + (long long)(mt * 16) * HDIM + nt * 16 + t;
    #pragma unroll
    for (int v = 0; v < 8; ++v)
        C[(long long)(v + 8 * g) * HDIM] = c[v];
}

// ------------- el/er: per-(p,n,head) dot over D=32 (one wave each) -------------
__global__ void k_attn_lr(const float* __restrict__ feat,
                          const float* __restrict__ al,
                          const float* __restrict__ ar,
                          float* __restrict__ el, float* __restrict__ er) {
    int wv   = blockIdx.x * (blockDim.x >> 5) + (threadIdx.x >> 5);
    int lane = threadIdx.x & 31;
    if (wv >= PP * NN * HH) return;
    int p   = wv / (NN * HH);
    int rem = wv % (NN * HH);
    int n   = rem / HH, hh = rem % HH;
    float f  = feat[((long long)p * NN + n) * HDIM + hh * DD + lane];
    float vl = f * al[((long long)p * HH + hh) * DD + lane];
    float vr = f * ar[((long long)p * HH + hh) * DD + lane];
    #pragma unroll
    for (int m = 1; m < 32; m <<= 1) {
        vl += __shfl_xor(vl, m, 32);
        vr += __shfl_xor(vr, m, 32);
    }
    if (lane == 0) { el[wv] = vl; er[wv] = vr; }
}

// ------------- segment max over dst -------------
__global__ void k_edge_max(const int* __restrict__ src, const int* __restrict__ dst,
                           const float* __restrict__ el, const float* __restrict__ er,
                           int* __restrict__ mmax) {
    long long i = blockIdx.x * (long long)blockDim.x + threadIdx.x;
    if (i >= (long long)PP * EE * HH) return;
    int hh = (int)(i % HH);
    long long pe = i / HH;                 // p*E + e
    int p = (int)(pe / EE);
    int s = src[pe], d = dst[pe];
    float ev = el[((long long)p * NN + s) * HH + hh] + er[((long long)p * NN + d) * HH + hh];
    ev = ev > 0.0f ? ev : NEG_SLOPE * ev;
    atomicMax(&mmax[((long long)p * NN + d) * HH + hh], enc_f(ev));
}

// ------------- ex = exp(e - m[dst]); denom = segment_sum(ex) -------------
__global__ void k_edge_exp(const int* __restrict__ src, const int* __restrict__ dst,
                           const float* __restrict__ el, const float* __restrict__ er,
                           const int* __restrict__ mmax,
                           float* __restrict__ exb, float* __restrict__ den) {
    long long i = blockIdx.x * (long long)blockDim.x + threadIdx.x;
    if (i >= (long long)PP * EE * HH) return;
    int hh = (int)(i % HH);
    long long pe = i / HH;
    int p = (int)(pe / EE);
    int s = src[pe], d = dst[pe];
    long long di = ((long long)p * NN + d) * HH + hh;
    float ev = el[((long long)p * NN + s) * HH + hh] + er[di];
    ev = ev > 0.0f ? ev : NEG_SLOPE * ev;
    float ex = __expf(ev - dec_i(mmax[di]));
    exb[i] = ex;
    atomicAdd(&den[di], ex);
}

// ------------- message scatter: acc[dst] += feat[src] * alpha -------------
__global__ void k_message(const int* __restrict__ src, const int* __restrict__ dst,
                          const float* __restrict__ feat,
                          const float* __restrict__ exb, const float* __restrict__ den,
                          float* __restrict__ acc) {
    long long i = blockIdx.x * (long long)blockDim.x + threadIdx.x;
    if (i >= (long long)PP * EE * HDIM) return;
    int c = (int)(i & (HDIM - 1));
    long long pe = i >> 7;                 // p*E + e
    int p = (int)(pe / EE);
    int s = src[pe], d = dst[pe];
    int hh = c >> 5;
    float alpha = exb[pe * HH + hh] / den[((long long)p * NN + d) * HH + hh];
    float val = feat[((long long)p * NN + s) * HDIM + c] * alpha;
    atomicAdd(&acc[((long long)p * NN + d) * HDIM + c], val);
}

// ------------- zp = elu(acc + bias) (in place) -------------
__global__ void k_bias_elu(float* __restrict__ acc, const float* __restrict__ bias) {
    long long i = blockIdx.x * (long long)blockDim.x + threadIdx.x;
    if (i >= (long long)PP * NN * HDIM) return;
    int c = (int)(i & (HDIM - 1));
    int p = (int)(i / ((long long)NN * HDIM));
    float x = acc[i] + bias[p * HDIM + c];
    acc[i] = x > 0.0f ? x : (__expf(x) - 1.0f);
}

// ------------- semantic attention GEMM: w[r] = tanh(zp@W1 + b1) @ W2 (WMMA) -------------
__global__ void k_sem_attn(const float* __restrict__ zp,
                           const float* __restrict__ W1,
                           const float* __restrict__ b1,
                           const float* __restrict__ W2,
                           float* __restrict__ wsem) {
    const int lane = threadIdx.x & 31;
    const int wid  = threadIdx.x >> 5;
    const int gid  = blockIdx.x * 8 + wid;
    const int MT = (PP * NN) / 16, NT = HIDS / 16;   // 9375, 8
    if (gid >= MT * NT) return;
    const int mt = gid / NT, nt = gid % NT;
    const int g = lane >> 4, t = lane & 15;

    const float* A = zp + (long long)(mt * 16 + t) * HDIM;
    const float* B = W1 + nt * 16 + t;

    v8f c = {};
    #pragma unroll 4
    for (int k = 0; k < HDIM; k += 4) {
        v2f a, b;
        a.x = A[k + 2 * g];
        a.y = A[k + 2 * g + 1];
        b.x = B[(long long)(k + 2 * g) * HIDS];
        b.y = B[(long long)(k + 2 * g + 1) * HIDS];
        c = __builtin_amdgcn_wmma_f32_16x16x4_f32(false, a, false, b,
                                                  (short)0, c, false, false);
    }
    const int col = nt * 16 + t;
    const float w2 = W2[col];
    const float bb = b1[col];
    #pragma unroll
    for (int v = 0; v < 8; ++v) {
        float val = tanhf(c[v] + bb) * w2;
        #pragma unroll
        for (int m = 1; m < 16; m <<= 1) val += __shfl_xor(val, m, 32);
        if (t == 0) atomicAdd(&wsem[mt * 16 + v + 8 * g], val);
    }
}

// ------------- reduce w over nodes per path -------------
__global__ void k_wsum(const float* __restrict__ wsem, float* __restrict__ red) {
    long long i = blockIdx.x * (long long)blockDim.x + threadIdx.x;
    if (i >= (long long)PP * NN) return;
    int p = (int)(i / NN);
    atomicAdd(&red[p], wsem[i]);
}

// ------------- beta = softmax(mean_n w) -------------
__global__ void k_beta(float* __restrict__ red) {
    if (threadIdx.x == 0 && blockIdx.x == 0) {
        float v[PP], m = -3.4e38f;
        for (int p = 0; p < PP; ++p) { v[p] = red[p] / (float)NN; m = fmaxf(m, v[p]); }
        float s = 0.0f;
        for (int p = 0; p < PP; ++p) { v[p] = __expf(v[p] - m); s += v[p]; }
        for (int p = 0; p < PP; ++p) red[4 + p] = v[p] / s;
    }
}

// ------------- final: out = (sum_p beta_p * zp_p) @ pred_W + pred_b -------------
__global__ void k_final(const float* __restrict__ zp, const float* __restrict__ red,
                        const float* __restrict__ pW, const float* __restrict__ pb,
                        float* __restrict__ out) {
    long long i = blockIdx.x * (long long)blockDim.x + threadIdx.x;
    if (i >= (long long)NN * OUTC) return;
    int n = (int)(i / OUTC), o = (int)(i % OUTC);
    float b0 = red[4], b1 = red[5], b2 = red[6];
    const float* z0 = zp + (long long)n * HDIM;
    const float* z1 = zp + ((long long)NN + n) * HDIM;
    const float* z2 = zp + ((long long)2 * NN + n) * HDIM;
    float acc = pb[o];
    #pragma unroll 8
    for (int c = 0; c < HDIM; ++c) {
        float z = b0 * z0[c] + b1 * z1[c] + b2 * z2[c];
        acc += z * pW[c * OUTC + o];
    }
    out[i] = acc;
}

extern "C" void kernel_launch(void* const* d_in, const int* in_sizes, int n_in,
                              void* d_out, int out_size, void* d_ws, size_t ws_size,
                              hipStream_t stream) {
    const float* h      = (const float*)d_in[0];
    const float* W      = (const float*)d_in[1];
    const float* attn_l = (const float*)d_in[2];
    const float* attn_r = (const float*)d_in[3];
    const float* bias   = (const float*)d_in[4];
    const float* sem_W1 = (const float*)d_in[5];
    const float* sem_b1 = (const float*)d_in[6];
    const float* sem_W2 = (const float*)d_in[7];
    const float* pred_W = (const float*)d_in[8];
    const float* pred_b = (const float*)d_in[9];
    const int*   src    = (const int*)d_in[10];
    const int*   dst    = (const int*)d_in[11];
    float* out = (float*)d_out;

    float* ws   = (float*)d_ws;
    float* feat = ws + OFF_FEAT;
    float* acc  = ws + OFF_ACC;
    float* el   = ws + OFF_EL;
    float* er   = ws + OFF_ER;
    int*   mmax = (int*)(ws + OFF_MM);
    float* den  = ws + OFF_DEN;
    float* exb  = ws + OFF_EXB;
    float* wsem = ws + OFF_WSE;
    float* red  = ws + OFF_RED;

    // 0) init scratch (acc=0, mmax=INT_MIN, den/wsem/red=0)
    k_init<<<4096, 256, 0, stream>>>(ws);

    // 1) feat = h @ W[p]  — 75000 waves, 9375 blocks of 8 waves
    k_gemm_feat<<<(PP * (NN / 16) * (HDIM / 16) + 7) / 8, 256, 0, stream>>>(h, W, feat);

    // 2) el/er — 600000 waves
    k_attn_lr<<<(PP * NN * HH + 7) / 8, 256, 0, stream>>>(feat, attn_l, attn_r, el, er);

    // 3) segment max — 9.6M threads
    long long eh = (long long)PP * EE * HH;
    k_edge_max<<<(unsigned)((eh + 255) / 256), 256, 0, stream>>>(src, dst, el, er, mmax);

    // 4) ex + denom
    k_edge_exp<<<(unsigned)((eh + 255) / 256), 256, 0, stream>>>(src, dst, el, er, mmax, exb, den);

    // 5) message scatter — 307.2M threads (bandwidth-dominant phase)
    long long mh = (long long)PP * EE * HDIM;
    k_message<<<(unsigned)((mh + 255) / 256), 256, 0, stream>>>(src, dst, feat, exb, den, acc);

    // 6) bias + ELU -> zp (in place)
    long long nh = (long long)PP * NN * HDIM;
    k_bias_elu<<<(unsigned)((nh + 255) / 256), 256, 0, stream>>>(acc, bias);

    // 7) semantic attention WMMA GEMM + tanh*W2 reduce
    k_sem_attn<<<(((PP * NN / 16) * (HIDS / 16)) + 7) / 8, 256, 0, stream>>>(
        acc, sem_W1, sem_b1, sem_W2, wsem);

    // 8) mean over nodes
    k_wsum<<<(PP * NN + 255) / 256, 256, 0, stream>>>(wsem, red);

    // 9) softmax over paths
    k_beta<<<1, 32, 0, stream>>>(red);

    // 10) weighted combine + prediction head
    k_final<<<(NN * OUTC + 255) / 256, 256, 0, stream>>>(acc, red, pred_W, pred_b, out);
}